// GNN_54949811585461
// MI455X (gfx1250) — compile-verified
//
#include <hip/hip_runtime.h>
#include <math.h>

typedef __attribute__((ext_vector_type(2))) float v2f;
typedef __attribute__((ext_vector_type(8))) float v8f;

#define N_NODES  100000
#define N_EDGES  1600000
#define N_LAYERS 3
#define HID      128
#define OUT      8
#define N_GRAPHS 128

#define ASTR 132   // LDS A-tile row stride (floats): 132%64==4 -> conflict-free 16x4 fragment loads, 8B aligned
#define KSTEPS (HID / 4)            // 32 WMMA k-steps
#define BPACK_PER_LAYER (8 * KSTEPS * 32)   // 8 column tiles * 32 ksteps * 32 lanes (float4 each)

// ---------------- zero scratch ----------------
__global__ void zero_f32(float* __restrict__ p, int n) {
    int i = blockIdx.x * blockDim.x + threadIdx.x;
    if (i < n) p[i] = 0.0f;
}

// ---------------- pack both weight matrices into per-lane WMMA B-fragment layout ----------------
// Bpack[tile nt][kstep ks][lane] = {Wrel[kk][n], Wrel[kk+1][n], Wroot[kk][n], Wroot[kk+1][n]}
// with koff = 2*(lane>=16), n = 16*nt + (lane&15), kk = 4*ks + koff.
// Inner GEMM loop then needs ONE coalesced global_load_b128 per k-step for all B data.
__global__ void pack_b(const float* __restrict__ Wrel, const float* __restrict__ Wroot,
                       float4* __restrict__ Bpack) {
    int i = blockIdx.x * blockDim.x + threadIdx.x;
    if (i >= BPACK_PER_LAYER) return;
    int lane = i & 31;
    int ks   = (i >> 5) & 31;
    int nt   = i >> 10;
    int koff = (lane >> 4) << 1;
    int n    = (nt << 4) + (lane & 15);
    int kk   = (ks << 2) + koff;
    float4 v;
    v.x = Wrel [kk       * HID + n];
    v.y = Wrel [(kk + 1) * HID + n];
    v.z = Wroot[kk       * HID + n];
    v.w = Wroot[(kk + 1) * HID + n];
    Bpack[i] = v;
}

// ---------------- edge scatter-add: agg[dst] += h[src] ----------------
// one wave32 per edge; each lane moves 4 consecutive floats (global_load_b128 + 4x global_atomic_add_f32)
__global__ void __launch_bounds__(256)
edge_scatter(const float* __restrict__ h, const int* __restrict__ src,
             const int* __restrict__ dst, float* __restrict__ agg, int n_edges) {
    int e    = (blockIdx.x * blockDim.x + threadIdx.x) >> 5;
    int lane = threadIdx.x & 31;
    if (e >= n_edges) return;
    int s = src[e];
    int d = dst[e];
    float4 v = *(const float4*)(h + (size_t)s * HID + lane * 4);
    float* ar = agg + (size_t)d * HID + lane * 4;
    unsafeAtomicAdd(ar + 0, v.x);
    unsafeAtomicAdd(ar + 1, v.y);
    unsafeAtomicAdd(ar + 2, v.z);
    unsafeAtomicAdd(ar + 3, v.w);
}

// ---------------- fused layer: hout = atan(agg@Wrel + hin@Wroot + brel) ----------------
// block = 256 thr (8 wave32) = one 16-node tile; wave w owns output cols [16w,16w+16)
// fp32 WMMA 16x16x4; both GEMMs fused into one accumulator; B from pre-packed fragments.
__global__ void __launch_bounds__(256)
gconv_layer(const float* __restrict__ agg, const float* __restrict__ hin,
            const float4* __restrict__ Bpack, const float* __restrict__ brel,
            float* __restrict__ hout) {
    __shared__ float sA[16 * ASTR];
    __shared__ float sH[16 * ASTR];

    const int tid  = threadIdx.x;
    const int wave = tid >> 5;
    const int lane = tid & 31;
    const size_t node0 = (size_t)blockIdx.x * 16;

    // cooperative stage of the two 16x128 A tiles into LDS (2 x float4 per thread each)
    {
        int r = tid >> 4;           // 0..15
        int c = (tid & 15) * 8;     // 0..120
        const float* ga = agg + (node0 + r) * HID + c;
        const float* gh = hin + (node0 + r) * HID + c;
        *(float4*)&sA[r * ASTR + c]     = *(const float4*)(ga);
        *(float4*)&sA[r * ASTR + c + 4] = *(const float4*)(ga + 4);
        *(float4*)&sH[r * ASTR + c]     = *(const float4*)(gh);
        *(float4*)&sH[r * ASTR + c + 4] = *(const float4*)(gh + 4);
    }
    __syncthreads();

    const int m    = lane & 15;            // A fragment row
    const int koff = (lane >> 4) << 1;     // 0 (lanes 0-15) or 2 (lanes 16-31)
    const int n    = (wave << 4) + m;      // global output column for this lane

    // C/D layout: VGPR r holds row (r + 8*(lane>=16)), col (lane&15). Bias depends only on n.
    v8f acc;
    {
        float bias = brel[n];
        #pragma unroll
        for (int r = 0; r < 8; ++r) acc[r] = bias;
    }

    // this wave's B-fragment stream: 32 consecutive float4 groups of 32 lanes
    const float4* Bp = Bpack + ((size_t)wave * KSTEPS * 32) + lane;

    #pragma unroll 4
    for (int ks = 0; ks < KSTEPS; ++ks) {
        const int k = ks << 2;
        v2f a  = *(const v2f*)&sA[m * ASTR + k + koff];   // ds_load_b64, bank-conflict-free
        v2f ah = *(const v2f*)&sH[m * ASTR + k + koff];
        float4 bb = Bp[ks * 32];                          // one global_load_b128, fully coalesced
        v2f b  = {bb.x, bb.y};
        v2f bq = {bb.z, bb.w};
        acc = __builtin_amdgcn_wmma_f32_16x16x4_f32(false, a,  false, b,  (short)0, acc, false, false);
        acc = __builtin_amdgcn_wmma_f32_16x16x4_f32(false, ah, false, bq, (short)0, acc, false, false);
    }

    const int rbase = (lane >> 4) << 3;    // 0 or 8
    #pragma unroll
    for (int r = 0; r < 8; ++r) {
        size_t row = node0 + rbase + r;
        hout[row * HID + n] = atanf(acc[r]);
    }
}

// ---------------- graph pooling: pooled[batch[node]] += h[node] ----------------
__global__ void __launch_bounds__(256)
pool_nodes(const float* __restrict__ h, const int* __restrict__ batch,
           float* __restrict__ pooled) {
    int node = (blockIdx.x * blockDim.x + threadIdx.x) >> 5;
    int lane = threadIdx.x & 31;
    if (node >= N_NODES) return;
    int g = batch[node];
    float4 v = *(const float4*)(h + (size_t)node * HID + lane * 4);
    float* pr = pooled + (size_t)g * HID + lane * 4;
    unsafeAtomicAdd(pr + 0, v.x);
    unsafeAtomicAdd(pr + 1, v.y);
    unsafeAtomicAdd(pr + 2, v.z);
    unsafeAtomicAdd(pr + 3, v.w);
}

// ---------------- head: out = sigmoid(pooled @ W_out + b_out) (tiny: 128x128x8) ----------------
__global__ void head(const float* __restrict__ pooled, const float* __restrict__ Wout,
                     const float* __restrict__ bout, float* __restrict__ out) {
    int i = blockIdx.x * blockDim.x + threadIdx.x;
    if (i >= N_GRAPHS * OUT) return;
    int g = i / OUT, o = i % OUT;
    float s = bout[o];
    #pragma unroll 4
    for (int k = 0; k < HID; ++k) s += pooled[g * HID + k] * Wout[k * OUT + o];
    out[i] = 1.0f / (1.0f + expf(-s));
}

extern "C" void kernel_launch(void* const* d_in, const int* in_sizes, int n_in,
                              void* d_out, int out_size, void* d_ws, size_t ws_size,
                              hipStream_t stream) {
    const float* x      = (const float*)d_in[0];
    const int*   ei     = (const int*)  d_in[1];   // [2, N_EDGES] (JAX x64 off -> int32)
    const int*   batch  = (const int*)  d_in[2];
    const float* W_rel  = (const float*)d_in[3];   // [3, 128, 128]
    const float* b_rel  = (const float*)d_in[4];   // [3, 128]
    const float* W_root = (const float*)d_in[5];   // [3, 128, 128]
    const float* W_out  = (const float*)d_in[6];   // [128, 8]
    const float* b_out  = (const float*)d_in[7];   // [8]
    float*       out    = (float*)d_out;

    const int* src = ei;
    const int* dst = ei + N_EDGES;

    float* h0     = (float*)d_ws;
    float* h1     = h0  + (size_t)N_NODES * HID;
    float* agg    = h1  + (size_t)N_NODES * HID;
    float* pooled = agg + (size_t)N_NODES * HID;
    float4* Bpack = (float4*)(pooled + (size_t)N_GRAPHS * HID);  // 3 * 128KB

    // repack weights into WMMA B-fragment order (once per call; tiny)
    for (int l = 0; l < N_LAYERS; ++l) {
        pack_b<<<(BPACK_PER_LAYER + 255) / 256, 256, 0, stream>>>(
            W_rel  + (size_t)l * HID * HID,
            W_root + (size_t)l * HID * HID,
            Bpack  + (size_t)l * BPACK_PER_LAYER);
    }

    const float* hin = x;
    float* bufs[2] = {h0, h1};

    for (int l = 0; l < N_LAYERS; ++l) {
        zero_f32<<<(N_NODES * HID + 255) / 256, 256, 0, stream>>>(agg, N_NODES * HID);
        edge_scatter<<<(N_EDGES * 32 + 255) / 256, 256, 0, stream>>>(hin, src, dst, agg, N_EDGES);
        float* hout = bufs[l & 1];
        gconv_layer<<<N_NODES / 16, 256, 0, stream>>>(
            agg, hin,
            Bpack + (size_t)l * BPACK_PER_LAYER,
            b_rel + (size_t)l * HID,
            hout);
        hin = hout;
    }

    zero_f32<<<(N_GRAPHS * HID + 255) / 256, 256, 0, stream>>>(pooled, N_GRAPHS * HID);
    pool_nodes<<<(N_NODES * 32 + 255) / 256, 256, 0, stream>>>(hin, batch, pooled);
    head<<<(N_GRAPHS * OUT + 255) / 256, 256, 0, stream>>>(pooled, W_out, b_out, out);
}